// MalwareDML_8718783610903
// MI455X (gfx1250) — compile-verified
//
#include <hip/hip_runtime.h>
#include <math.h>

#define NN      100000      // N_NODES
#define NE      1600000     // N_EDGES
#define NG      512         // N_GRAPHS
#define FIN     128
#define H1D     128
#define EMBD    256
#define ATTH    64

typedef float v2f __attribute__((ext_vector_type(2)));
typedef float v8f __attribute__((ext_vector_type(8)));

static __host__ int cdiv(long long a, long long b) { return (int)((a + b - 1) / b); }

// ---------------------------------------------------------------------------
// WMMA fp32 GEMM:  C[M,N] = A[M,K] @ B[K,N] (+ bias[N]) (+ relu)
// One wave32 per 16x64 output strip (4 N-tiles): the A fragment is loaded once
// per K-step and reused by 4 V_WMMA_F32_16X16X4_F32 issues -> 4 independent
// accumulator chains (ILP for the matrix pipe) and 1.6x better L2 intensity.
// Lane layout per ISA 7.12.2 (32-bit A 16x4 / B 4x16 / C 16x16).
// Requires M%16==0, N%64==0, K%4==0 (true for all uses here).
// ---------------------------------------------------------------------------
template <bool RELU>
__global__ void gemm_wmma_f32(const float* __restrict__ A,
                              const float* __restrict__ B,
                              const float* __restrict__ bias,
                              float* __restrict__ C,
                              int M, int N, int K)
{
    const int wavesPerBlock = blockDim.x >> 5;
    const int wid    = blockIdx.x * wavesPerBlock + (threadIdx.x >> 5);
    const int strips = N >> 6;                    // 16x64 strips per row band
    const int tilesM = M >> 4;
    if (wid >= tilesM * strips) return;           // uniform per-wave exit
    const int tm = wid / strips;
    const int ts = wid - tm * strips;

    const int lane = threadIdx.x & 31;
    const int half = lane >> 4;                   // 0 -> K {0,1}; 1 -> K {2,3}
    const int l16  = lane & 15;
    const int m    = (tm << 4) + l16;             // A row held by this lane
    const int n0   = (ts << 6) + l16;             // first B/C column held by this lane

    const float* __restrict__ Arow = A + (size_t)m * K;
    v8f acc0 = {}, acc1 = {}, acc2 = {}, acc3 = {};
    for (int k = 0; k < K; k += 4) {
        const int ka = k + (half << 1);
        v2f a = *(const v2f*)(Arow + ka);         // A[m][ka], A[m][ka+1]
        const float* __restrict__ Brow0 = B + (size_t)ka * N + n0;
        const float* __restrict__ Brow1 = Brow0 + N;
        v2f b0, b1, b2, b3;
        b0.x = Brow0[0];  b0.y = Brow1[0];        // B[ka][n], B[ka+1][n]
        b1.x = Brow0[16]; b1.y = Brow1[16];
        b2.x = Brow0[32]; b2.y = Brow1[32];
        b3.x = Brow0[48]; b3.y = Brow1[48];
        acc0 = __builtin_amdgcn_wmma_f32_16x16x4_f32(false, a, false, b0, (short)0, acc0, false, false);
        acc1 = __builtin_amdgcn_wmma_f32_16x16x4_f32(false, a, false, b1, (short)0, acc1, false, false);
        acc2 = __builtin_amdgcn_wmma_f32_16x16x4_f32(false, a, false, b2, (short)0, acc2, false, false);
        acc3 = __builtin_amdgcn_wmma_f32_16x16x4_f32(false, a, false, b3, (short)0, acc3, false, false);
    }

    const int rbase = (tm << 4) + (half << 3);    // lanes 0-15: rows 0-7; 16-31: rows 8-15
    v8f accs[4] = {acc0, acc1, acc2, acc3};
#pragma unroll
    for (int t = 0; t < 4; ++t) {
        const int n  = n0 + (t << 4);
        const float bv = (bias != nullptr) ? bias[n] : 0.0f;
#pragma unroll
        for (int j = 0; j < 8; ++j) {
            float v = accs[t][j] + bv;
            if (RELU) v = fmaxf(v, 0.0f);
            C[(size_t)(rbase + j) * N + n] = v;
        }
    }
}

// ---------------------------------------------------------------------------
// Degree / normalization helpers (GCN with self loops: deg = in_deg + 1)
// ---------------------------------------------------------------------------
__global__ void k_deg_init(float* __restrict__ deg)
{
    int i = blockIdx.x * blockDim.x + threadIdx.x;
    if (i < NN) deg[i] = 1.0f;                    // self loop
}

__global__ void k_deg_edges(const int* __restrict__ dst, float* __restrict__ deg)
{
    int e = blockIdx.x * blockDim.x + threadIdx.x;
    if (e < NE) atomicAdd(&deg[dst[e]], 1.0f);
}

__global__ void k_dinv(const float* __restrict__ deg, float* __restrict__ dinv)
{
    int i = blockIdx.x * blockDim.x + threadIdx.x;
    if (i < NN) dinv[i] = rsqrtf(deg[i]);         // deg >= 1 always
}

__global__ void k_edge_norm(const int* __restrict__ src, const int* __restrict__ dst,
                            const float* __restrict__ dinv, float* __restrict__ nrm)
{
    int e = blockIdx.x * blockDim.x + threadIdx.x;
    if (e < NE) nrm[e] = dinv[src[e]] * dinv[dst[e]];
}

// out[n][:] = h[n][:] * dinv[n]^2   (self-loop contribution, initializes out)
__global__ void k_self_init(const float* __restrict__ h, const float* __restrict__ dinv,
                            float* __restrict__ out, int F4)
{
    long long tid   = (long long)blockIdx.x * blockDim.x + threadIdx.x;
    long long total = (long long)NN * F4;
    if (tid >= total) return;
    int nid = (int)(tid / F4);
    int c   = (int)(tid - (long long)nid * F4);
    float s = dinv[nid]; s *= s;
    float4 hv = *(const float4*)(h + ((size_t)nid * F4 + c) * 4);
    float4 ov = make_float4(hv.x * s, hv.y * s, hv.z * s, hv.w * s);
    *(float4*)(out + ((size_t)nid * F4 + c) * 4) = ov;
}

// out[dst[e]][:] += h[src[e]][:] * nrm[e]   (4 features per thread)
__global__ void k_edge_scatter(const float* __restrict__ h,
                               const int* __restrict__ src, const int* __restrict__ dst,
                               const float* __restrict__ nrm,
                               float* __restrict__ out, int F4)
{
    long long tid   = (long long)blockIdx.x * blockDim.x + threadIdx.x;
    long long total = (long long)NE * F4;
    if (tid >= total) return;
    int e = (int)(tid / F4);
    int c = (int)(tid - (long long)e * F4);
    int s = src[e], d = dst[e];
    float w = nrm[e];
    float4 hv = *(const float4*)(h + ((size_t)s * F4 + c) * 4);
    float* o  = out + ((size_t)d * F4 + c) * 4;
    atomicAdd(o + 0, hv.x * w);
    atomicAdd(o + 1, hv.y * w);
    atomicAdd(o + 2, hv.z * w);
    atomicAdd(o + 3, hv.w * w);
}

// io[n][f] = relu(io[n][f] + bias[f])            (conv1 epilogue, in place)
__global__ void k_epi_relu_bias(float* __restrict__ io, const float* __restrict__ bias, int F4)
{
    long long tid   = (long long)blockIdx.x * blockDim.x + threadIdx.x;
    long long total = (long long)NN * F4;
    if (tid >= total) return;
    int c = (int)(tid % F4);
    float4 v = *(float4*)(io + tid * 4);
    const float4 b = *(const float4*)(bias + c * 4);
    v.x = fmaxf(v.x + b.x, 0.0f);
    v.y = fmaxf(v.y + b.y, 0.0f);
    v.z = fmaxf(v.z + b.z, 0.0f);
    v.w = fmaxf(v.w + b.w, 0.0f);
    *(float4*)(io + tid * 4) = v;
}

// io[n][f] = relu(io[n][f] + bias[f] + add[n][f])   (h2 epilogue; add = h1@Wres+bres)
__global__ void k_epi_h2(float* __restrict__ io, const float* __restrict__ bias,
                         const float* __restrict__ add, int F4)
{
    long long tid   = (long long)blockIdx.x * blockDim.x + threadIdx.x;
    long long total = (long long)NN * F4;
    if (tid >= total) return;
    int c = (int)(tid % F4);
    float4 v = *(float4*)(io + tid * 4);
    const float4 b = *(const float4*)(bias + c * 4);
    const float4 a = *(const float4*)(add + tid * 4);
    v.x = fmaxf(v.x + b.x + a.x, 0.0f);
    v.y = fmaxf(v.y + b.y + a.y, 0.0f);
    v.z = fmaxf(v.z + b.z + a.z, 0.0f);
    v.w = fmaxf(v.w + b.w + a.w, 0.0f);
    *(float4*)(io + tid * 4) = v;
}

// w[n] = sigmoid( relu_att[n][:] . Wa2 + ba2 )
__global__ void k_node_w(const float* __restrict__ att, const float* __restrict__ Wa2,
                         const float* __restrict__ ba2, float* __restrict__ w)
{
    int nid = blockIdx.x * blockDim.x + threadIdx.x;
    if (nid >= NN) return;
    const float* a = att + (size_t)nid * ATTH;
    float s = ba2[0];
#pragma unroll
    for (int i = 0; i < ATTH; ++i) s += a[i] * Wa2[i];
    w[nid] = 1.0f / (1.0f + __expf(-s));
}

__global__ void k_zero(float* __restrict__ p, int n)
{
    int i = blockIdx.x * blockDim.x + threadIdx.x;
    if (i < n) p[i] = 0.0f;
}

// esum[batch[n]][f] += h2[n][f] * w[n]
__global__ void k_pool(const float* __restrict__ h2, const float* __restrict__ w,
                       const int* __restrict__ batch, float* __restrict__ esum)
{
    long long tid   = (long long)blockIdx.x * blockDim.x + threadIdx.x;
    long long total = (long long)NN * (EMBD / 4);
    if (tid >= total) return;
    int nid = (int)(tid / (EMBD / 4));
    int c   = (int)(tid - (long long)nid * (EMBD / 4));
    int g   = batch[nid];
    float wv = w[nid];
    float4 hv = *(const float4*)(h2 + ((size_t)nid * (EMBD / 4) + c) * 4);
    float* o  = esum + ((size_t)g * (EMBD / 4) + c) * 4;
    atomicAdd(o + 0, hv.x * wv);
    atomicAdd(o + 1, hv.y * wv);
    atomicAdd(o + 2, hv.z * wv);
    atomicAdd(o + 3, hv.w * wv);
}

__global__ void k_pool_cnt(const int* __restrict__ batch, float* __restrict__ cnt)
{
    int nid = blockIdx.x * blockDim.x + threadIdx.x;
    if (nid < NN) atomicAdd(&cnt[batch[nid]], 1.0f);
}

// emb = (esum/max(cnt,1)); out = emb / max(||emb||_2, 1e-12).  1 block per graph.
__global__ void k_finalize(const float* __restrict__ esum, const float* __restrict__ cnt,
                           float* __restrict__ out)
{
    __shared__ float red[EMBD];
    int g = blockIdx.x, f = threadIdx.x;
    float c = fmaxf(cnt[g], 1.0f);
    float v = esum[(size_t)g * EMBD + f] / c;
    red[f] = v * v;
    __syncthreads();
    for (int s = EMBD / 2; s > 0; s >>= 1) {
        if (f < s) red[f] += red[f + s];
        __syncthreads();
    }
    float nrm = fmaxf(sqrtf(red[0]), 1e-12f);
    out[(size_t)g * EMBD + f] = v / nrm;
}

// ---------------------------------------------------------------------------
// Launch
// ---------------------------------------------------------------------------
extern "C" void kernel_launch(void* const* d_in, const int* in_sizes, int n_in,
                              void* d_out, int out_size, void* d_ws, size_t ws_size,
                              hipStream_t stream)
{
    (void)in_sizes; (void)n_in; (void)out_size; (void)ws_size;

    const float* x    = (const float*)d_in[0];
    const int*   ei   = (const int*)d_in[1];
    const int*   src  = ei;
    const int*   dst  = ei + NE;
    const int*   batch= (const int*)d_in[2];
    const float* W1   = (const float*)d_in[3];
    const float* b1   = (const float*)d_in[4];
    const float* W2   = (const float*)d_in[5];
    const float* b2   = (const float*)d_in[6];
    const float* Wres = (const float*)d_in[7];
    const float* bres = (const float*)d_in[8];
    const float* Wa1  = (const float*)d_in[9];
    const float* ba1  = (const float*)d_in[10];
    const float* Wa2  = (const float*)d_in[11];
    const float* ba2  = (const float*)d_in[12];
    float* out = (float*)d_out;

    // ---- workspace layout (bytes); buffers reused across phases ----
    char* ws = (char*)d_ws;
    float* T1   = (float*)(ws + 0);                 //  51.2 MB  X@W1         (dead after conv1 scatter)
    float* H1   = (float*)(ws + 51200000ULL);       //  51.2 MB  agg1 -> h1 (in place)
    float* T2   = (float*)(ws + 102400000ULL);      // 102.4 MB  H1@W2        (dead after conv2 scatter)
    float* T3   = (float*)(ws + 204800000ULL);      // 102.4 MB  H1@Wres+bres (dead after h2 epilogue)
    float* H2   = (float*)(ws + 307200000ULL);      // 102.4 MB  agg2 -> h2 (in place)
    float* DEG  = (float*)(ws + 409600000ULL);      //   0.4 MB
    float* DINV = (float*)(ws + 410000000ULL);      //   0.4 MB
    float* NRM  = (float*)(ws + 410400000ULL);      //   6.4 MB
    float* ESUM = (float*)(ws + 416800000ULL);      // 512*256*4
    float* CNT  = (float*)(ws + 417324288ULL);      // 512*4
    float* ATT  = (float*)(ws + 0);                 //  25.6 MB  (reuses T1 region, live after T1 dies)
    float* WN   = (float*)(ws + 25600000ULL);       //   0.4 MB  node weights (reuses T1 region)

    const int TB = 256;

    // 1) degrees + symmetric normalization
    k_deg_init <<<cdiv(NN, TB), TB, 0, stream>>>(DEG);
    k_deg_edges<<<cdiv(NE, TB), TB, 0, stream>>>(dst, DEG);
    k_dinv     <<<cdiv(NN, TB), TB, 0, stream>>>(DEG, DINV);
    k_edge_norm<<<cdiv(NE, TB), TB, 0, stream>>>(src, dst, DINV, NRM);

    // 2) conv1: T1 = X @ W1 ; agg ; H1 = relu(agg + b1)
    {
        int waves = (NN / 16) * (H1D / 64);
        gemm_wmma_f32<false><<<cdiv(waves, 4), 128, 0, stream>>>(x, W1, nullptr, T1, NN, H1D, FIN);
    }
    k_self_init   <<<cdiv((long long)NN * (H1D / 4), TB), TB, 0, stream>>>(T1, DINV, H1, H1D / 4);
    k_edge_scatter<<<cdiv((long long)NE * (H1D / 4), TB), TB, 0, stream>>>(T1, src, dst, NRM, H1, H1D / 4);
    k_epi_relu_bias<<<cdiv((long long)NN * (H1D / 4), TB), TB, 0, stream>>>(H1, b1, H1D / 4);

    // 3) conv2 + residual: T2 = H1@W2 ; T3 = H1@Wres + bres ; agg ; H2 = relu(agg + b2 + T3)
    {
        int waves = (NN / 16) * (EMBD / 64);
        gemm_wmma_f32<false><<<cdiv(waves, 4), 128, 0, stream>>>(H1, W2,   nullptr, T2, NN, EMBD, H1D);
        gemm_wmma_f32<false><<<cdiv(waves, 4), 128, 0, stream>>>(H1, Wres, bres,    T3, NN, EMBD, H1D);
    }
    k_self_init   <<<cdiv((long long)NN * (EMBD / 4), TB), TB, 0, stream>>>(T2, DINV, H2, EMBD / 4);
    k_edge_scatter<<<cdiv((long long)NE * (EMBD / 4), TB), TB, 0, stream>>>(T2, src, dst, NRM, H2, EMBD / 4);
    k_epi_h2      <<<cdiv((long long)NN * (EMBD / 4), TB), TB, 0, stream>>>(H2, b2, T3, EMBD / 4);

    // 4) attention: ATT = relu(H2 @ Wa1 + ba1) ; WN = sigmoid(ATT . Wa2 + ba2)
    {
        int waves = (NN / 16) * (ATTH / 64);   // one 16x64 strip covers N=64
        gemm_wmma_f32<true><<<cdiv(waves, 4), 128, 0, stream>>>(H2, Wa1, ba1, ATT, NN, ATTH, EMBD);
    }
    k_node_w<<<cdiv(NN, TB), TB, 0, stream>>>(ATT, Wa2, ba2, WN);

    // 5) weighted mean pool per graph + L2 normalize
    k_zero    <<<cdiv(NG * EMBD, TB), TB, 0, stream>>>(ESUM, NG * EMBD);
    k_zero    <<<cdiv(NG, TB), TB, 0, stream>>>(CNT, NG);
    k_pool    <<<cdiv((long long)NN * (EMBD / 4), TB), TB, 0, stream>>>(H2, WN, batch, ESUM);
    k_pool_cnt<<<cdiv(NN, TB), TB, 0, stream>>>(batch, CNT);
    k_finalize<<<NG, EMBD, 0, stream>>>(ESUM, CNT, out);
}